// LocallyConnected1d_21998822490173
// MI455X (gfx1250) — compile-verified
//
#include <hip/hip_runtime.h>

typedef __attribute__((ext_vector_type(2))) float v2f;
typedef __attribute__((ext_vector_type(4))) float v4f;
typedef __attribute__((ext_vector_type(8))) float v8f;

#define B_     32
#define CIN    128
#define COUT   128
#define L_     2048
#define K_     3
#define W_     2050            // (L-1)*1 + 3
#define LT     8               // l positions per block
#define OB     64              // outputs per block
#define CC     4               // channels per chunk
#define CKC    (CC * K_)       // 12: K-extent per chunk
#define NCHUNK (CIN / CC)      // 32
#define WSTRIDE 14             // even (8B-aligned b64), 14*n mod 64 conflict-free
#define ASTRIDE 14
#define WADV   (CC * L_ * K_)  // 24576 floats per chunk
#define XADV   (CC * W_)       // 8200 floats per chunk

// wlds: [li][o][ck]  8*64*14 floats = 28,672 B   (B fragments)
// alds: [li][b][ck]  8*32*14 floats = 14,336 B   (A fragments, im2col-expanded)
#define WLDS_LI (OB * WSTRIDE)      // 896 floats per li plane
#define ALDS_LI (B_ * ASTRIDE)      // 448 floats per li plane

struct Regs {
    v4f   wq[6];    // one (o,c) 24-float weight run
    float xv[5];    // 5 x samples (this half's wloc range)
};

__device__ __forceinline__ void load_regs(Regs& r, const float* wp, const float* xp) {
#pragma unroll
    for (int q = 0; q < 6; ++q) r.wq[q] = *(const v4f*)(wp + q * 4);   // b128 clause
#pragma unroll
    for (int u = 0; u < 5; ++u) r.xv[u] = xp[u];
}

__device__ __forceinline__ void store_lds(const Regs& r, float* wst, float* ast, int half) {
#pragma unroll
    for (int q = 0; q < 6; ++q) {
#pragma unroll
        for (int j = 0; j < 4; ++j) {
            const int lk = q * 4 + j;                 // unroll-constant
            const int li = lk / 3;
            const int k  = lk - li * 3;
            wst[li * WLDS_LI + k] = r.wq[q][j];       // immediate-offset ds_store
        }
    }
    if (half == 0) {   // wloc 0..4  (wave-uniform branch)
        ast[0 * ALDS_LI + 0] = r.xv[0];
        ast[1 * ALDS_LI + 0] = r.xv[1]; ast[0 * ALDS_LI + 1] = r.xv[1];
        ast[2 * ALDS_LI + 0] = r.xv[2]; ast[1 * ALDS_LI + 1] = r.xv[2]; ast[0 * ALDS_LI + 2] = r.xv[2];
        ast[3 * ALDS_LI + 0] = r.xv[3]; ast[2 * ALDS_LI + 1] = r.xv[3]; ast[1 * ALDS_LI + 2] = r.xv[3];
        ast[4 * ALDS_LI + 0] = r.xv[4]; ast[3 * ALDS_LI + 1] = r.xv[4]; ast[2 * ALDS_LI + 2] = r.xv[4];
    } else {           // wloc 5..9
        ast[5 * ALDS_LI + 0] = r.xv[0]; ast[4 * ALDS_LI + 1] = r.xv[0]; ast[3 * ALDS_LI + 2] = r.xv[0];
        ast[6 * ALDS_LI + 0] = r.xv[1]; ast[5 * ALDS_LI + 1] = r.xv[1]; ast[4 * ALDS_LI + 2] = r.xv[1];
        ast[7 * ALDS_LI + 0] = r.xv[2]; ast[6 * ALDS_LI + 1] = r.xv[2]; ast[5 * ALDS_LI + 2] = r.xv[2];
                                        ast[7 * ALDS_LI + 1] = r.xv[3]; ast[6 * ALDS_LI + 2] = r.xv[3];
                                                                        ast[7 * ALDS_LI + 2] = r.xv[4];
    }
}

__device__ __forceinline__ void compute_chunk(v8f (&acc)[LT], const float* aF, const float* bF) {
#pragma unroll
    for (int li = 0; li < LT; ++li) {
#pragma unroll
        for (int kk3 = 0; kk3 < CKC / 4; ++kk3) {
            v2f av = *(const v2f*)(aF + li * ALDS_LI + kk3 * 4);  // ds_load_b64, imm offset
            v2f bv = *(const v2f*)(bF + li * WLDS_LI + kk3 * 4);  // ds_load_b64, imm offset
            acc[li] = __builtin_amdgcn_wmma_f32_16x16x4_f32(
                false, av, false, bv, (short)0, acc[li], false, false);
        }
    }
}

__global__ __launch_bounds__(256)
void lc1d_wmma_f32(const float* __restrict__ x,
                   const float* __restrict__ w,
                   const float* __restrict__ bias,
                   float* __restrict__ out)
{
    __shared__ __align__(16) float wlds[LT * WLDS_LI];
    __shared__ __align__(16) float alds[LT * ALDS_LI];

    const int tid  = threadIdx.x;
    const int lane = tid & 31;
    const int wave = tid >> 5;
    const int l0   = blockIdx.x * LT;
    const int o0   = blockIdx.y * OB;

    const int hi   = lane >> 4;
    const int lo16 = lane & 15;
    const int m0   = (wave >> 2) * 16;   // batch tile base: 0 or 16
    const int n0   = (wave & 3) * 16;    // local output tile base

    // ---- per-thread loader bases ----
    const int lw_c = tid & 3;
    const int lw_o = tid >> 2;
    const float* wp = w + ((size_t)(o0 + lw_o) * CIN + lw_c) * (L_ * K_) + (size_t)l0 * K_;
    float* wst = &wlds[lw_o * WSTRIDE + lw_c * K_];

    const int lx_c = (tid & 127) & 3;
    const int lx_b = (tid & 127) >> 2;
    const int half = tid >> 7;                            // wave-uniform
    const float* xp = x + ((size_t)lx_b * CIN + lx_c) * W_ + l0 + half * 5;
    float* ast = &alds[lx_b * ASTRIDE + lx_c * K_];

    // ---- per-lane compute bases ----
    const float* aF = &alds[(m0 + lo16) * ASTRIDE + 2 * hi];
    const float* bF = &wlds[(n0 + lo16) * WSTRIDE + 2 * hi];

    v8f acc[LT];
#pragma unroll
    for (int i = 0; i < LT; ++i)
        acc[i] = (v8f){0.f, 0.f, 0.f, 0.f, 0.f, 0.f, 0.f, 0.f};

    // ---- software pipeline: regs double-buffer, 2x-unrolled chunk loop ----
    Regs rA, rB;
    load_regs(rA, wp, xp);               // chunk 0
    wp += WADV; xp += XADV;              // -> chunk 1

#pragma unroll 1
    for (int ch = 0; ch < NCHUNK; ch += 2) {
        __syncthreads();                          // LDS free (prev compute done)
        store_lds(rA, wst, ast, half);
        load_regs(rB, wp, xp);                    // prefetch chunk ch+1 (always valid)
        wp += WADV; xp += XADV;
        __syncthreads();                          // LDS visible
        compute_chunk(acc, aF, bF);               // overlaps rB global loads

        __syncthreads();
        store_lds(rB, wst, ast, half);
        if (ch + 2 < NCHUNK) {                    // prefetch chunk ch+2 (skip on last)
            load_regs(rA, wp, xp);
            wp += WADV; xp += XADV;
        }
        __syncthreads();
        compute_chunk(acc, aF, bF);               // overlaps rA global loads
    }

    // ---- epilogue: fused bias add, non-temporal b128 stores ----
    const int o_glob = o0 + n0 + lo16;
    const float* bp  = bias + (size_t)o_glob * L_ + l0;
    v4f bias0 = *(const v4f*)bp;
    v4f bias1 = *(const v4f*)(bp + 4);
#pragma unroll
    for (int r = 0; r < 8; ++r) {
        const int m = m0 + r + 8 * hi;            // batch index per D-matrix layout
        float* op = out + ((size_t)m * COUT + o_glob) * L_ + l0;
        v4f v0, v1;
#pragma unroll
        for (int q = 0; q < 4; ++q) { v0[q] = acc[q][r]     + bias0[q]; }
#pragma unroll
        for (int q = 0; q < 4; ++q) { v1[q] = acc[q + 4][r] + bias1[q]; }
        __builtin_nontemporal_store(v0, (v4f*)op);        // never re-read
        __builtin_nontemporal_store(v1, (v4f*)(op + 4));
    }
}

extern "C" void kernel_launch(void* const* d_in, const int* in_sizes, int n_in,
                              void* d_out, int out_size, void* d_ws, size_t ws_size,
                              hipStream_t stream) {
    const float* x    = (const float*)d_in[0];   // (32, 128, 2050)
    const float* wgt  = (const float*)d_in[1];   // (128, 128, 2048, 3)
    const float* bias = (const float*)d_in[2];   // (128, 2048)
    float* out        = (float*)d_out;           // (32, 128, 2048)

    dim3 grid(L_ / LT, COUT / OB);               // (256, 2)
    dim3 block(256);
    lc1d_wmma_f32<<<grid, block, 0, stream>>>(x, wgt, bias, out);
}